// graph_atn_47845935677671
// MI455X (gfx1250) — compile-verified
//
#include <hip/hip_runtime.h>
#include <hip/hip_bf16.h>

#define DK 32
#define ODIM 32

typedef __attribute__((ext_vector_type(2))) float v2f;
typedef __attribute__((ext_vector_type(8))) float v8f;
typedef unsigned int __attribute__((ext_vector_type(4))) u32x4;
typedef int __attribute__((ext_vector_type(4))) i32x4;
typedef int __attribute__((ext_vector_type(8))) i32x8;

// ---- float <-> order-preserving unsigned encoding (for atomic max) ----
__device__ __forceinline__ unsigned f2ord(float f) {
    int i = __float_as_int(f);
    return (i >= 0) ? ((unsigned)i | 0x80000000u) : (unsigned)(~i);
}
__device__ __forceinline__ float ord2f(unsigned u) {
    int i = (u & 0x80000000u) ? (int)(u & 0x7fffffffu) : ~(int)u;
    return __int_as_float(i);
}

// ---- zero init of segmax/denom/agg (contiguous region of u32 words) ----
__global__ void k_zero(unsigned* p, int n) {
    int i = blockIdx.x * blockDim.x + threadIdx.x;
    if (i < n) p[i] = 0u;
}

// ---- pass 1: per-edge score + per-segment max (wave per edge, lane = dim) ----
__global__ void k_scores(const float* __restrict__ X, const float* __restrict__ K,
                         const int* __restrict__ src, const int* __restrict__ dst,
                         float* __restrict__ scores, unsigned* __restrict__ segmax, int E) {
    int lane  = threadIdx.x & 31;
    int wave  = blockIdx.x * (blockDim.x >> 5) + (threadIdx.x >> 5);
    int nwv   = gridDim.x * (blockDim.x >> 5);
    for (int e = wave; e < E; e += nwv) {
        int s = src[e];
        int d = dst[e];
        float p = X[(size_t)d * DK + lane] * K[(size_t)s * DK + lane];
        #pragma unroll
        for (int off = 16; off; off >>= 1) p += __shfl_xor(p, off, 32);
        p *= (1.0f / (float)DK);
        if (lane == 0) {
            scores[e] = p;
            atomicMax(&segmax[d], f2ord(p));
        }
    }
}

// ---- pass 2: e = exp(score - max); denom[dst] += e ----
__global__ void k_expsum(const int* __restrict__ dst, float* __restrict__ scores,
                         const unsigned* __restrict__ segmax, float* __restrict__ denom, int E) {
    int e = blockIdx.x * blockDim.x + threadIdx.x;
    if (e >= E) return;
    __builtin_prefetch(dst + e + 8192, 0, 0);   // gfx1250 global_prefetch
    int d   = dst[e];
    float m = ord2f(segmax[d]);
    float ev = __expf(scores[e] - m);
    scores[e] = ev;                              // reuse buffer for exp values
    atomicAdd(&denom[d], ev);
}

// ---- pass 3: agg[dst] += alpha * V[src]; run-length-compressed atomics ----
__global__ void k_agg(const float* __restrict__ V,
                      const int* __restrict__ src, const int* __restrict__ dst,
                      const float* __restrict__ expv, const float* __restrict__ denom,
                      float* __restrict__ agg, int E, int epw) {
    int lane = threadIdx.x & 31;
    int wave = blockIdx.x * (blockDim.x >> 5) + (threadIdx.x >> 5);
    long e0 = (long)wave * epw;
    if (e0 >= E) return;
    int eend = (int)((e0 + epw < (long)E) ? e0 + epw : (long)E);
    int cur  = dst[(int)e0];
    float acc = 0.0f;
    for (int e = (int)e0; e < eend; ++e) {
        int d = dst[e];                          // uniform across wave
        if (d != cur) {                          // uniform branch
            atomicAdd(&agg[(size_t)cur * DK + lane], acc);
            acc = 0.0f;
            cur = d;
        }
        float alpha = expv[e] / denom[d];
        acc += alpha * V[(size_t)src[e] * DK + lane];
    }
    atomicAdd(&agg[(size_t)cur * DK + lane], acc);
}

// ---- pass 4: out = agg @ W_o + b using V_WMMA_F32_16X16X4_F32 ----
// One wave per 16-row tile; two 16x16 accumulators (ODIM=32).
// W_o is staged into LDS via the Tensor Data Mover (1-D tile, 1024 f32).
// A 16x4 f32 layout : lanes 0-15 -> M=lane, K={k0,k0+1}; lanes 16-31 -> K={k0+2,k0+3}
// B 4x16 f32 layout : lanes 0-15 -> K={k0,k0+1}, N=lane; lanes 16-31 -> K={k0+2,k0+3}
// C/D 16x16 f32     : VGPR r -> rows r (lanes 0-15) / r+8 (lanes 16-31)
__global__ void k_proj(const float* __restrict__ agg, const float* __restrict__ Wo,
                       const float* __restrict__ bo, float* __restrict__ out, int N) {
    __shared__ float ldsW[DK * ODIM];   // 4 KB

    int lane = threadIdx.x;             // 32 threads/block: one wave, EXEC all-1s
    int half = lane >> 4;               // 0 or 1
    int l15  = lane & 15;

    // --- TDM: async DMA W_o (32x32 f32, contiguous) into LDS ---
    {
        unsigned lds_off = (unsigned)(size_t)(void*)ldsW;     // low 32 bits = LDS byte offset
        unsigned long long ga = (unsigned long long)(const void*)Wo;
        // D# group 0: count=1 | lds_addr | global_addr | type=2
        u32x4 g0;
        g0[0] = 1u;
        g0[1] = lds_off;
        g0[2] = (unsigned)(ga & 0xffffffffu);
        g0[3] = (unsigned)((ga >> 32) & 0x1ffffffu) | (2u << 30);
        // D# group 1: data_size=4B; tensor_dim0=1024; tensor_dim1=1;
        //             tile_dim0=1024; tile_dim1=1; tensor_dim0_stride=1024
        i32x8 g1;
        g1[0] = (2 << 16);              // data_size = 4 bytes
        g1[1] = (int)(1024u << 16);     // tensor_dim0[15:0] in bits [63:48]
        g1[2] = (1 << 16);              // tensor_dim0 hi16=0 | tensor_dim1[15:0]=1
        g1[3] = (int)(1024u << 16);     // tensor_dim1 hi16=0 | tile_dim0=1024
        g1[4] = 1;                      // tile_dim1=1 | tile_dim2=0
        g1[5] = 1024;                   // tensor_dim0_stride lo32
        g1[6] = (int)(1024u << 16);     // stride hi16=0 | tensor_dim1_stride lo16
        g1[7] = 0;
        i32x4 gz = {0, 0, 0, 0};
#if defined(__clang_major__) && (__clang_major__ >= 23)
        i32x8 gz8 = {0, 0, 0, 0, 0, 0, 0, 0};
        __builtin_amdgcn_tensor_load_to_lds(g0, g1, gz, gz, gz8, 0);
#else
        __builtin_amdgcn_tensor_load_to_lds(g0, g1, gz, gz, 0);
#endif
        __builtin_amdgcn_s_wait_tensorcnt(0);   // single wave: no barrier needed
    }

    // A row pointer: clamp OOB rows (their WMMA outputs are never stored)
    int row = blockIdx.x * 16 + l15;
    row = (row < N) ? row : (N - 1);
    const float* arow = agg + (size_t)row * DK;

    v8f acc0 = {};
    v8f acc1 = {};
    #pragma unroll
    for (int k0 = 0; k0 < DK; k0 += 4) {
        int kk = k0 + half * 2;
        v2f a = *(const v2f*)(arow + kk);       // one global_load_b64, no branches
        v2f b0, b1;
        b0.x = ldsW[kk * ODIM + l15];
        b0.y = ldsW[(kk + 1) * ODIM + l15];
        b1.x = ldsW[kk * ODIM + 16 + l15];
        b1.y = ldsW[(kk + 1) * ODIM + 16 + l15];
        acc0 = __builtin_amdgcn_wmma_f32_16x16x4_f32(false, a, false, b0,
                                                     (short)0, acc0, false, false);
        acc1 = __builtin_amdgcn_wmma_f32_16x16x4_f32(false, a, false, b1,
                                                     (short)0, acc1, false, false);
    }

    float bias0 = bo[l15];
    float bias1 = bo[16 + l15];
    int orow0 = blockIdx.x * 16 + half * 8;
    #pragma unroll
    for (int r = 0; r < 8; ++r) {
        int orow = orow0 + r;
        if (orow < N) {                          // divergence only after all WMMAs
            out[(size_t)orow * ODIM + l15]      = acc0[r] + bias0;
            out[(size_t)orow * ODIM + 16 + l15] = acc1[r] + bias1;
        }
    }
}

extern "C" void kernel_launch(void* const* d_in, const int* in_sizes, int n_in,
                              void* d_out, int out_size, void* d_ws, size_t ws_size,
                              hipStream_t stream) {
    const float* X   = (const float*)d_in[0];
    const float* K   = (const float*)d_in[1];
    const float* V   = (const float*)d_in[2];
    const float* Wo  = (const float*)d_in[3];
    const float* bo  = (const float*)d_in[4];
    const int*   src = (const int*)d_in[5];
    const int*   dst = (const int*)d_in[6];
    float* out = (float*)d_out;

    int N = in_sizes[0] / DK;   // 100000
    int E = in_sizes[5];        // 1600000

    // workspace layout: [scores E f32][segmax N u32][denom N f32][agg N*32 f32]
    char* ws = (char*)d_ws;
    size_t off = 0;
    float*    scores = (float*)(ws + off);     off += ((size_t)E * 4 + 255) & ~(size_t)255;
    unsigned* segmax = (unsigned*)(ws + off);  off += (size_t)N * 4;
    float*    denom  = (float*)(ws + off);     off += (size_t)N * 4;
    float*    agg    = (float*)(ws + off);     off += (size_t)N * DK * 4;

    // 1) zero segmax + denom + agg (contiguous, 34*N words)
    int zero_words = N * (2 + DK);
    k_zero<<<(zero_words + 255) / 256, 256, 0, stream>>>(segmax, zero_words);

    // 2) scores + segment max (wave per edge)
    k_scores<<<2048, 256, 0, stream>>>(X, K, src, dst, scores, segmax, E);

    // 3) exp + denom
    k_expsum<<<(E + 255) / 256, 256, 0, stream>>>(dst, scores, segmax, denom, E);

    // 4) aggregation (wave per chunk of 64 edges, run-length compressed atomics)
    const int EPW = 64;
    int nwaves = (E + EPW - 1) / EPW;
    int nblocks = (nwaves + 7) / 8;  // 8 waves per 256-thread block
    k_agg<<<nblocks, 256, 0, stream>>>(V, src, dst, scores, denom, agg, E, EPW);

    // 5) output projection with f32 WMMA (W_o staged via TDM)
    k_proj<<<(N + 15) / 16, 32, 0, stream>>>(agg, Wo, bo, out, N);
}